// TropicalVisionTransformer_47614007443914
// MI455X (gfx1250) — compile-verified
//
#include <hip/hip_runtime.h>

// Tropical (max,+) Vision Transformer for MI455X (gfx1250).
// Matrix ops are max-plus: WMMA cannot express them, so the CDNA5-specific
// path here is the Tensor Data Mover: every weight tile is DMA'd into LDS
// with TDM row padding (+2 DWORDs/row) so compute-phase LDS reads are
// bank-conflict-free, synchronized with s_wait_tensorcnt.

#define NB   8      // batch
#define IMG  224
#define PP   16     // patch side
#define GG   14     // patches per side
#define NP   196    // patches (tokens)
#define PD   256    // patch dim
#define DM   128    // model dim
#define DF   256    // ff dim
#define NC   1000   // classes
#define RPB  14     // token rows per block (NP / GG)
#define NEG_INF (-3.0e38f)

typedef unsigned int u32;
typedef __attribute__((ext_vector_type(4))) u32 u32x4;
typedef __attribute__((ext_vector_type(8))) int i32x8;
typedef __attribute__((ext_vector_type(4))) int i32x4;

// ---------------------------------------------------------------------------
// TDM: DMA a rows x cols f32 tile (row stride = stride_elems in memory) into
// LDS at lds_dst, padding each row so the LDS pitch becomes cols + 2 floats.
// pad_code: pad every 2^(pad_code+1) DWORDs; amt_code: by (amt_code+1) DWORDs.
// Issue from ONE wave only; EXEC is ignored by tensor ops.
// ---------------------------------------------------------------------------
__device__ __forceinline__ void tdm_load_2d(const float* gsrc, float* lds_dst,
                                            u32 rows, u32 cols, u32 stride_elems,
                                            u32 pad_code, u32 amt_code)
{
    unsigned long long ga = (unsigned long long)(size_t)gsrc;
    u32 lds_addr = (u32)(size_t)(void*)lds_dst;   // low 32 bits = LDS offset

    u32x4 g0;
    g0[0] = 1u;                                   // count=1, user descriptor
    g0[1] = lds_addr;                             // lds_addr [63:32]
    g0[2] = (u32)ga;                              // global_addr low
    g0[3] = ((u32)(ga >> 32) & 0x01FFFFFFu)       // global_addr[56:32]
            | 0x80000000u;                        // type=2 ("image")

    i32x8 g1;
    g1[0] = (int)((2u << 16)                      // data_size = 4 bytes
                | (1u << 20)                      // pad_enable
                | (pad_code << 22)                // pad_interval
                | (amt_code << 25));              // pad_amount
    g1[1] = (int)((cols & 0xFFFFu) << 16);        // tensor_dim0 low16 (abar=0)
    g1[2] = (int)(((cols >> 16) & 0xFFFFu) | ((rows & 0xFFFFu) << 16)); // dim0 hi | dim1 lo
    g1[3] = (int)(((rows >> 16) & 0xFFFFu) | ((cols & 0xFFFFu) << 16)); // dim1 hi | tile_dim0
    g1[4] = (int)(rows & 0xFFFFu);                // tile_dim1 (tile_dim2 = 0)
    g1[5] = (int)stride_elems;                    // tensor_dim0_stride low32
    g1[6] = 0;                                    // stride hi / dim1_stride lo
    g1[7] = 0;

    i32x4 gz4 = {0, 0, 0, 0};                     // groups 2/3 unused (2D)
    i32x8 gz8 = {0, 0, 0, 0, 0, 0, 0, 0};         // extra arg (zero-filled)
    __builtin_amdgcn_tensor_load_to_lds(g0, g1, gz4, gz4, gz8, 0);
}

// Block-wide max over n participating threads (red has n slots).
__device__ __forceinline__ float block_max(float v, float* red, int t, int n)
{
    red[t] = v;
    __syncthreads();
    for (int s = n >> 1; s > 0; s >>= 1) {
        if (t < s) red[t] = fmaxf(red[t], red[t + s]);
        __syncthreads();
    }
    float m = red[0];
    __syncthreads();
    return m;
}

// ---------------------------------------------------------------------------
// 1) patchify + tropical embed + positional add: h[b,n,d] = max_j(p+W) + pos
// ---------------------------------------------------------------------------
__global__ void __launch_bounds__(DM) k_embed(const float* __restrict__ x,
                                              const float* __restrict__ eW,
                                              const float* __restrict__ pos,
                                              float* __restrict__ h)
{
    extern __shared__ float smem[];
    float* Wl = smem;                 // 128 x 258 (padded pitch)
    float* xs = Wl + DM * 258;        // 256
    const int t  = threadIdx.x;       // 0..127
    const int b  = blockIdx.x / GG;
    const int n0 = (blockIdx.x % GG) * RPB;

    if (t < 32) {
        tdm_load_2d(eW, Wl, DM, PD, PD, 7u, 1u);   // pad 2 DW every 256 DW
        __builtin_amdgcn_s_wait_tensorcnt(0);
    }
    __syncthreads();

    const float2* xs2 = (const float2*)xs;
    for (int r = 0; r < RPB; ++r) {
        const int n = n0 + r, gy = n / GG, gx = n % GG;
        {   // gather one 16x16 patch row into LDS (2 floats / thread)
            const int j = 2 * t, py = j >> 4, px = j & 15;
            const float2 v2 = *(const float2*)(x + (size_t)((b * IMG + gy * PP + py) * IMG + gx * PP + px));
            *(float2*)(xs + j) = v2;
        }
        __syncthreads();
        float m = NEG_INF;
        const float2* wr = (const float2*)(Wl + t * 258);
        #pragma unroll 8
        for (int j = 0; j < PD / 2; ++j) {
            float2 w = wr[j], xv = xs2[j];
            m = fmaxf(m, fmaxf(xv.x + w.x, xv.y + w.y));
        }
        h[((size_t)(b * NP + n)) * DM + t] = m + pos[n * DM + t];
        __syncthreads();
    }
}

// ---------------------------------------------------------------------------
// 2) q,k,v = trop_mm(pnorm(h), W): fused as max_j(h+W) - rowmax(h)
// ---------------------------------------------------------------------------
__global__ void __launch_bounds__(DM) k_qkv(const float* __restrict__ h,
                                            const float* __restrict__ qW,
                                            const float* __restrict__ kW,
                                            const float* __restrict__ vW,
                                            float* __restrict__ q,
                                            float* __restrict__ k,
                                            float* __restrict__ v)
{
    extern __shared__ float smem[];
    float* qw = smem;                  // 128 x 130 each
    float* kw = qw + DM * 130;
    float* vw = kw + DM * 130;
    float* xs = vw + DM * 130;         // 128
    float* red = xs + DM;              // 128
    const int t  = threadIdx.x;
    const int b  = blockIdx.x / GG;
    const int n0 = (blockIdx.x % GG) * RPB;

    if (t < 32) {
        tdm_load_2d(qW, qw, DM, DM, DM, 6u, 1u);   // pad 2 DW every 128 DW
        tdm_load_2d(kW, kw, DM, DM, DM, 6u, 1u);
        tdm_load_2d(vW, vw, DM, DM, DM, 6u, 1u);
        __builtin_amdgcn_s_wait_tensorcnt(0);
    }
    __syncthreads();

    const float2* xs2 = (const float2*)xs;
    for (int r = 0; r < RPB; ++r) {
        const size_t base = ((size_t)(b * NP + n0 + r)) * DM;
        float xv = h[base + t];
        xs[t] = xv;
        float rm = block_max(xv, red, t, DM);
        float mq = NEG_INF, mk = NEG_INF, mv = NEG_INF;
        const float2* qr = (const float2*)(qw + t * 130);
        const float2* kr = (const float2*)(kw + t * 130);
        const float2* vr = (const float2*)(vw + t * 130);
        #pragma unroll 8
        for (int j = 0; j < DM / 2; ++j) {
            float2 xj = xs2[j], a = qr[j], c = kr[j], d = vr[j];
            mq = fmaxf(mq, fmaxf(xj.x + a.x, xj.y + a.y));
            mk = fmaxf(mk, fmaxf(xj.x + c.x, xj.y + c.y));
            mv = fmaxf(mv, fmaxf(xj.x + d.x, xj.y + d.y));
        }
        q[base + t] = mq - rm;
        k[base + t] = mk - rm;
        v[base + t] = mv - rm;
        __syncthreads();
    }
}

// ---------------------------------------------------------------------------
// 3) tropical attention + pnorm + residual (in-place on h).
//    Score rowmax subtraction cancels exactly in (max,+), so it is skipped.
// ---------------------------------------------------------------------------
__global__ void __launch_bounds__(DM) k_attn(const float* __restrict__ q,
                                             const float* __restrict__ k,
                                             const float* __restrict__ v,
                                             float* __restrict__ h)
{
    extern __shared__ float smem[];
    float* kl   = smem;                // 196 x 130
    float* vl   = kl + NP * 130;       // 196 x 130
    float* qrow = vl + NP * 130;       // 128
    float* sc   = qrow + DM;           // 200 (196 used)
    float* red  = sc + 200;            // 128
    const int t  = threadIdx.x;
    const int b  = blockIdx.x / GG;
    const int i0 = (blockIdx.x % GG) * RPB;

    if (t < 32) {
        tdm_load_2d(k + (size_t)b * NP * DM, kl, NP, DM, DM, 6u, 1u);
        tdm_load_2d(v + (size_t)b * NP * DM, vl, NP, DM, DM, 6u, 1u);
        __builtin_amdgcn_s_wait_tensorcnt(0);
    }
    __syncthreads();

    const float2* q2 = (const float2*)qrow;
    const float2* sc2 = (const float2*)sc;
    for (int r = 0; r < RPB; ++r) {
        const size_t base = ((size_t)(b * NP + i0 + r)) * DM;
        qrow[t] = q[base + t];
        __syncthreads();
        for (int jj = t; jj < NP; jj += DM) {       // scores for rows jj
            float s = NEG_INF;
            const float2* kr = (const float2*)(kl + jj * 130);
            #pragma unroll 8
            for (int d = 0; d < DM / 2; ++d) {
                float2 qd = q2[d], kd = kr[d];
                s = fmaxf(s, fmaxf(qd.x + kd.x, qd.y + kd.y));
            }
            sc[jj] = s;
        }
        __syncthreads();
        float m = NEG_INF;                          // out_d = max_j(sc_j + v_jd)
        #pragma unroll 7
        for (int j2 = 0; j2 < NP / 2; ++j2) {
            float2 s2 = sc2[j2];
            m = fmaxf(m, fmaxf(s2.x + vl[(2 * j2) * 130 + t],
                               s2.y + vl[(2 * j2 + 1) * 130 + t]));
        }
        float amax = block_max(m, red, t, DM);      // pnorm over d
        h[base + t] = fmaxf(h[base + t], m - amax); // tropical residual
        __syncthreads();
    }
}

// ---------------------------------------------------------------------------
// 4) fused FFN: hmid = max(trop_mm(pnorm(h), ff1W), tau) (row-local, in LDS);
//               h    = max(h, pnorm(trop_mm(hmid, ff2W)))
//    Both weight tiles resident in LDS (~262 KB of the 320 KB/WGP).
// ---------------------------------------------------------------------------
__global__ void __launch_bounds__(DF) k_ffn(const float* __restrict__ h_in,
                                            const float* __restrict__ f1W,
                                            const float* __restrict__ f2W,
                                            const float* __restrict__ tau,
                                            float* __restrict__ h)
{
    extern __shared__ float smem[];
    float* w1 = smem;                  // 256 x 130
    float* w2 = w1 + DF * 130;         // 128 x 258
    float* hm = w2 + DM * 258;         // 256
    float* xs = hm + DF;               // 128
    float* red = xs + DM;              // 256
    const int t  = threadIdx.x;        // 0..255
    const int b  = blockIdx.x / GG;
    const int n0 = (blockIdx.x % GG) * RPB;
    const float tv = tau[0];

    if (t < 32) {
        tdm_load_2d(f1W, w1, DF, DM, DM, 6u, 1u);  // pad 2 DW every 128 DW
        tdm_load_2d(f2W, w2, DM, DF, DF, 7u, 1u);  // pad 2 DW every 256 DW
        __builtin_amdgcn_s_wait_tensorcnt(0);
    }
    __syncthreads();

    const float2* xs2 = (const float2*)xs;
    const float2* hm2 = (const float2*)hm;
    for (int r = 0; r < RPB; ++r) {
        const size_t base = ((size_t)(b * NP + n0 + r)) * DM;
        float xv = NEG_INF;
        if (t < DM) { xv = h_in[base + t]; xs[t] = xv; }
        float rm = block_max(xv, red, t, DF);

        float m1 = NEG_INF;                         // hidden: 256 outputs
        const float2* w1r = (const float2*)(w1 + t * 130);
        #pragma unroll 8
        for (int j = 0; j < DM / 2; ++j) {
            float2 xj = xs2[j], w = w1r[j];
            m1 = fmaxf(m1, fmaxf(xj.x + w.x, xj.y + w.y));
        }
        hm[t] = fmaxf(m1 - rm, tv);                 // tropical ReLU vs tau
        __syncthreads();

        float m2 = NEG_INF;                         // back to 128 outputs
        if (t < DM) {
            const float2* w2r = (const float2*)(w2 + t * 258);
            #pragma unroll 8
            for (int j = 0; j < DF / 2; ++j) {
                float2 hj = hm2[j], w = w2r[j];
                m2 = fmaxf(m2, fmaxf(hj.x + w.x, hj.y + w.y));
            }
        }
        float mx = block_max(m2, red, t, DF);       // pnorm over d
        if (t < DM) h[base + t] = fmaxf(h_in[base + t], m2 - mx);
        __syncthreads();
    }
}

// ---------------------------------------------------------------------------
// 5) tropical global pool over tokens + tropical classifier head * scale
// ---------------------------------------------------------------------------
__global__ void __launch_bounds__(256) k_head(const float* __restrict__ h,
                                              const float* __restrict__ hW,
                                              const float* __restrict__ lscale,
                                              float* __restrict__ out)
{
    __shared__ float pool[DM];
    const int t = threadIdx.x;         // 0..255
    const int b = blockIdx.x;
    if (t < DM) {
        float m = NEG_INF;
        for (int n = 0; n < NP; ++n) m = fmaxf(m, h[((size_t)(b * NP + n)) * DM + t]);
        pool[t] = m;
    }
    __syncthreads();
    const float ls = lscale[0];
    const float2* p2 = (const float2*)pool;
    for (int c = t; c < NC; c += 256) {
        const float2* wr = (const float2*)(hW + (size_t)c * DM);
        float m = NEG_INF;
        #pragma unroll 8
        for (int d = 0; d < DM / 2; ++d) {
            float2 pd = p2[d], w = wr[d];
            m = fmaxf(m, fmaxf(pd.x + w.x, pd.y + w.y));
        }
        out[(size_t)b * NC + c] = m * ls;
    }
}

// ---------------------------------------------------------------------------
extern "C" void kernel_launch(void* const* d_in, const int* in_sizes, int n_in,
                              void* d_out, int out_size, void* d_ws, size_t ws_size,
                              hipStream_t stream)
{
    (void)in_sizes; (void)n_in; (void)out_size; (void)ws_size;
    const float* x     = (const float*)d_in[0];
    const float* eW    = (const float*)d_in[1];
    const float* pos   = (const float*)d_in[2];
    const float* headW = (const float*)d_in[15];
    const float* lsc   = (const float*)d_in[16];
    float* out = (float*)d_out;

    float* h  = (float*)d_ws;                     // 8*196*128
    float* q  = h  + (size_t)NB * NP * DM;
    float* k  = q  + (size_t)NB * NP * DM;
    float* v  = k  + (size_t)NB * NP * DM;

    const int grid = NB * GG;                     // 112 workgroups

    k_embed<<<grid, DM, (DM * 258 + PD) * sizeof(float), stream>>>(x, eW, pos, h);
    for (int l = 0; l < 2; ++l) {
        const float* qW  = (const float*)d_in[3 + 6 * l + 0];
        const float* kW  = (const float*)d_in[3 + 6 * l + 1];
        const float* vW  = (const float*)d_in[3 + 6 * l + 2];
        const float* f1W = (const float*)d_in[3 + 6 * l + 3];
        const float* f2W = (const float*)d_in[3 + 6 * l + 4];
        const float* tau = (const float*)d_in[3 + 6 * l + 5];
        k_qkv <<<grid, DM, (3 * DM * 130 + 2 * DM) * sizeof(float), stream>>>(h, qW, kW, vW, q, k, v);
        k_attn<<<grid, DM, (2 * NP * 130 + 2 * DM + 200) * sizeof(float), stream>>>(q, k, v, h);
        k_ffn <<<grid, DF, (DF * 130 + DM * 258 + DF + DM + DF) * sizeof(float), stream>>>(h, f1W, f2W, tau, h);
    }
    k_head<<<NB, 256, 0, stream>>>(h, headW, lsc, out);
}